// BANLayer_32736240730180
// MI455X (gfx1250) — compile-verified
//
#include <hip/hip_runtime.h>
#include <math.h>

// Problem constants (from reference)
#define BB    32
#define VN    512
#define QN    1024
#define DDIM  128
#define HK    384
#define HOUT  4
#define HD    128

typedef float v2f __attribute__((ext_vector_type(2)));
typedef float v8f __attribute__((ext_vector_type(8)));

static __device__ __forceinline__ v8f wmma_f32x4(v2f a, v2f b, v8f c) {
  // D(16x16,f32) = A(16x4,f32) * B(4x16,f32) + C
  return __builtin_amdgcn_wmma_f32_16x16x4_f32(
      /*neg_a=*/false, a, /*neg_b=*/false, b,
      /*c_mod=*/(short)0, c, /*reuse_a=*/false, /*reuse_b=*/false);
}

static __device__ __forceinline__ v2f ld2(const float* p) {
  return *(const v2f*)p;
}

// ---------------------------------------------------------------------------
// Sinusoidal positional encoding table: pe[pos, d]
// ---------------------------------------------------------------------------
__global__ void pe_kernel(float* __restrict__ pe, int L) {
  int pos = blockIdx.x;
  int d   = threadIdx.x;
  int i   = d >> 1;
  float div = expf(-(float)(2 * i) * (logf(10000.0f) / (float)DDIM));
  float ang = (float)pos * div;
  pe[pos * DDIM + d] = (d & 1) ? cosf(ang) : sinf(ang);
}

// ---------------------------------------------------------------------------
// Fused prep GEMMs: Y1 = relu((X+PE)@W1 + b1), Y2 = relu(X@W2 + b2)
// X: (Mtot x 128) row-major, W: (128 x 384) row-major, Y: (Mtot x 384).
// Each wave: one 16(M) x 32(N) tile of BOTH outputs. 8 waves / block.
// ---------------------------------------------------------------------------
__global__ void prep_kernel(const float* __restrict__ X, const float* __restrict__ PE,
                            const float* __restrict__ W1, const float* __restrict__ b1,
                            const float* __restrict__ W2, const float* __restrict__ b2,
                            float* __restrict__ Y1, float* __restrict__ Y2,
                            int Npos) {
  const int warp = threadIdx.x >> 5;
  const int lane = threadIdx.x & 31;
  const int r    = lane & 15;
  const int half = lane >> 4;
  const int w    = blockIdx.x * 8 + warp;
  const int nt   = w % 12;          // 12 n-tiles of 32 cols
  const int mt   = w / 12;
  const int m0   = mt * 16;
  const int n0   = nt * 32;

  const int row   = m0 + r;
  const int perow = row & (Npos - 1);   // Npos is a power of two

  v8f acc1[2], acc2[2];
  const v8f vzero = {0.f, 0.f, 0.f, 0.f, 0.f, 0.f, 0.f, 0.f};
#pragma unroll
  for (int s = 0; s < 2; s++) { acc1[s] = vzero; acc2[s] = vzero; }

  const float* xrow = X  + (size_t)row   * DDIM;
  const float* prow = PE + (size_t)perow * DDIM;

  for (int kk = 0; kk < DDIM; kk += 4) {
    const int kc = kk + 2 * half;         // this half-wave's K columns
    v2f xa = ld2(xrow + kc);
    v2f pa = ld2(prow + kc);
    v2f a1 = xa + pa;                     // PE branch
    v2f a2 = xa;                          // raw branch
#pragma unroll
    for (int s = 0; s < 2; s++) {
      const int col = n0 + 16 * s + r;
      v2f bf1, bf2;
      bf1[0] = W1[(size_t)(kc    ) * HK + col];
      bf1[1] = W1[(size_t)(kc + 1) * HK + col];
      bf2[0] = W2[(size_t)(kc    ) * HK + col];
      bf2[1] = W2[(size_t)(kc + 1) * HK + col];
      acc1[s] = wmma_f32x4(a1, bf1, acc1[s]);
      acc2[s] = wmma_f32x4(a2, bf2, acc2[s]);
    }
  }

#pragma unroll
  for (int s = 0; s < 2; s++) {
    const int col = n0 + 16 * s + r;
    const float bia1 = b1[col];
    const float bia2 = b2[col];
#pragma unroll
    for (int j = 0; j < 8; j++) {
      const int orow = m0 + j + 8 * half;
      float y1 = acc1[s][j] + bia1; y1 = y1 > 0.f ? y1 : 0.f;
      float y2 = acc2[s][j] + bia2; y2 = y2 > 0.f ? y2 : 0.f;
      Y1[(size_t)orow * HK + col] = y1;
      Y2[(size_t)orow * HK + col] = y2;
    }
  }
}

// ---------------------------------------------------------------------------
// att[b,h,v,q] = sum_k h_mat[h,k]*v_[b,v,k]*q_[b,q,k] + h_bias[h]
// Per wave: 16(v) x 64(q) tile for one (b,h).  8 waves / block.
// ---------------------------------------------------------------------------
__global__ void att_kernel(const float* __restrict__ Vh, const float* __restrict__ Qh,
                           const float* __restrict__ h_mat, const float* __restrict__ h_bias,
                           float* __restrict__ att) {
  const int warp = threadIdx.x >> 5;
  const int lane = threadIdx.x & 31;
  const int r    = lane & 15;
  const int half = lane >> 4;
  const int w    = blockIdx.x * 8 + warp;   // 65536 wave-tiles total
  const int nt   = w & 15;                  // 16 n-tiles of 64
  const int mt   = (w >> 4) & 31;           // 32 m-tiles of 16
  const int bh   = w >> 9;                  // 0..127
  const int b    = bh >> 2;
  const int h    = bh & 3;
  const int m0   = mt * 16;
  const int n0   = nt * 64;

  const float hb = h_bias[h];
  v8f acc[4];
  const v8f cinit = {hb, hb, hb, hb, hb, hb, hb, hb};
#pragma unroll
  for (int s = 0; s < 4; s++) acc[s] = cinit;

  const float* vrow  = Vh + (size_t)(b * VN + m0 + r) * HK;
  const float* hrow  = h_mat + (size_t)h * HK;
  const float* qbase = Qh + (size_t)b * QN * HK;

  for (int kk = 0; kk < HK; kk += 4) {
    const int kc = kk + 2 * half;
    v2f a  = ld2(vrow + kc);
    v2f hm = ld2(hrow + kc);
    a = a * hm;                             // fold h_mat scale into A fragment
#pragma unroll
    for (int s = 0; s < 4; s++) {
      v2f bf = ld2(qbase + (size_t)(n0 + 16 * s + r) * HK + kc);
      acc[s] = wmma_f32x4(a, bf, acc[s]);
    }
  }

  const size_t obase = ((size_t)(b * HOUT + h)) * VN * QN;
#pragma unroll
  for (int s = 0; s < 4; s++) {
    const int col = n0 + 16 * s + r;
#pragma unroll
    for (int j = 0; j < 8; j++) {
      const int orow = m0 + j + 8 * half;
      att[obase + (size_t)orow * QN + col] = acc[s][j];
    }
  }
}

// ---------------------------------------------------------------------------
// Per-batch Gram matrix: G[b,k,k'] = sum_v Y[b,v,k] * Z[b,v,k']
// Per wave: 16(k) x 64(k') tile. 144 wave-tiles per batch, 8 waves / block.
// ---------------------------------------------------------------------------
__global__ void corr_kernel(const float* __restrict__ Y, const float* __restrict__ Z,
                            float* __restrict__ G, int Nrows) {
  const int warp = threadIdx.x >> 5;
  const int lane = threadIdx.x & 31;
  const int r    = lane & 15;
  const int half = lane >> 4;
  const int w    = blockIdx.x * 8 + warp;
  const int nt   = w % 6;                  // 6 n-tiles of 64
  const int mt   = (w / 6) % 24;           // 24 m-tiles of 16
  const int b    = w / 144;
  const int m0   = mt * 16;
  const int n0   = nt * 64;

  const float* Yb = Y + (size_t)b * Nrows * HK;
  const float* Zb = Z + (size_t)b * Nrows * HK;

  v8f acc[4];
  const v8f vzero = {0.f, 0.f, 0.f, 0.f, 0.f, 0.f, 0.f, 0.f};
#pragma unroll
  for (int s = 0; s < 4; s++) acc[s] = vzero;

  for (int vv = 0; vv < Nrows; vv += 4) {
    const int vr = vv + 2 * half;
    v2f a;                                   // A[m,kk] = Y[vr+.., m0+r] (transposed read)
    a[0] = Yb[(size_t)(vr    ) * HK + m0 + r];
    a[1] = Yb[(size_t)(vr + 1) * HK + m0 + r];
#pragma unroll
    for (int s = 0; s < 4; s++) {
      const int col = n0 + 16 * s + r;
      v2f bf;
      bf[0] = Zb[(size_t)(vr    ) * HK + col];
      bf[1] = Zb[(size_t)(vr + 1) * HK + col];
      acc[s] = wmma_f32x4(a, bf, acc[s]);
    }
  }

#pragma unroll
  for (int s = 0; s < 4; s++) {
    const int col = n0 + 16 * s + r;
#pragma unroll
    for (int j = 0; j < 8; j++) {
      const int orow = m0 + j + 8 * half;
      G[(size_t)b * HK * HK + (size_t)orow * HK + col] = acc[s][j];
    }
  }
}

// ---------------------------------------------------------------------------
// Column sums: S[b,k] = sum_v Y[b,v,k]
// ---------------------------------------------------------------------------
__global__ void colsum_kernel(const float* __restrict__ Y, float* __restrict__ S, int Nrows) {
  int t = blockIdx.x * blockDim.x + threadIdx.x;
  if (t >= BB * HK) return;
  int b = t / HK, k = t % HK;
  const float* p = Y + (size_t)b * Nrows * HK + k;
  float s = 0.f;
  for (int v = 0; v < Nrows; v++) s += p[(size_t)v * HK];
  S[t] = s;
}

// Hsum[k] = sum_h h_mat[h,k]; hbsum = sum_h h_bias[h]
__global__ void hsum_kernel(const float* __restrict__ h_mat, const float* __restrict__ h_bias,
                            float* __restrict__ Hsum, float* __restrict__ hbsum) {
  int k = threadIdx.x;
  if (k < HK) {
    float s = 0.f;
    for (int h = 0; h < HOUT; h++) s += h_mat[(size_t)h * HK + k];
    Hsum[k] = s;
  }
  if (k == 0) {
    float s = 0.f;
    for (int h = 0; h < HOUT; h++) s += h_bias[h];
    *hbsum = s;
  }
}

// ---------------------------------------------------------------------------
// logits_k[b,k] = sum_k' Hsum[k']*A[b,k,k']*B[b,k,k'] + hbsum*Sv[b,k]*Sq[b,k]
// One block (128 thr) per (b,k).
// ---------------------------------------------------------------------------
__global__ void logitsk_kernel(const float* __restrict__ A, const float* __restrict__ Bm,
                               const float* __restrict__ Hsum, const float* __restrict__ hbsum,
                               const float* __restrict__ Sv, const float* __restrict__ Sq,
                               float* __restrict__ Lk) {
  const int bk = blockIdx.x;            // b*HK + k
  const int b  = bk / HK;
  const int k  = bk % HK;
  const float* Ar = A  + (size_t)b * HK * HK + (size_t)k * HK;
  const float* Br = Bm + (size_t)b * HK * HK + (size_t)k * HK;
  __shared__ float red[128];
  float p = 0.f;
  for (int kp = threadIdx.x; kp < HK; kp += 128)
    p += Hsum[kp] * Ar[kp] * Br[kp];
  red[threadIdx.x] = p;
  __syncthreads();
  for (int off = 64; off > 0; off >>= 1) {
    if (threadIdx.x < off) red[threadIdx.x] += red[threadIdx.x + off];
    __syncthreads();
  }
  if (threadIdx.x == 0)
    Lk[bk] = red[0] + (*hbsum) * Sv[bk] * Sq[bk];
}

// ---------------------------------------------------------------------------
// logits[b,d] = sum_j Lk[b,3d+j]; then BatchNorm over batch (biased var).
// One block (32 thr == batch) per feature d.
// ---------------------------------------------------------------------------
__global__ void final_kernel(const float* __restrict__ Lk,
                             const float* __restrict__ gamma, const float* __restrict__ beta,
                             float* __restrict__ out) {
  const int d = blockIdx.x;   // 0..127
  const int b = threadIdx.x;  // 0..31
  float v = Lk[(size_t)b * HK + 3 * d] + Lk[(size_t)b * HK + 3 * d + 1] +
            Lk[(size_t)b * HK + 3 * d + 2];
  __shared__ float sh[32];
  sh[b] = v;
  __syncthreads();
  float mu = 0.f;
  for (int i = 0; i < 32; i++) mu += sh[i];
  mu *= (1.f / 32.f);
  __syncthreads();
  float dv = v - mu;
  sh[b] = dv * dv;
  __syncthreads();
  float var = 0.f;
  for (int i = 0; i < 32; i++) var += sh[i];
  var *= (1.f / 32.f);
  out[(size_t)b * HD + d] = dv * rsqrtf(var + 1e-5f) * gamma[d] + beta[d];
}

// ---------------------------------------------------------------------------
extern "C" void kernel_launch(void* const* d_in, const int* in_sizes, int n_in,
                              void* d_out, int out_size, void* d_ws, size_t ws_size,
                              hipStream_t stream) {
  const float* v      = (const float*)d_in[0];
  const float* q      = (const float*)d_in[1];
  const float* Wv     = (const float*)d_in[2];
  const float* bv     = (const float*)d_in[3];
  const float* Wq     = (const float*)d_in[4];
  const float* bq     = (const float*)d_in[5];
  const float* Wvv    = (const float*)d_in[6];
  const float* bvv    = (const float*)d_in[7];
  const float* Wvq    = (const float*)d_in[8];
  const float* bvq    = (const float*)d_in[9];
  const float* h_mat  = (const float*)d_in[10];
  const float* h_bias = (const float*)d_in[11];
  const float* gamma  = (const float*)d_in[12];
  const float* beta   = (const float*)d_in[13];

  float* out_logits = (float*)d_out;                 // (32,128)
  float* out_att    = (float*)d_out + BB * HD;       // (32,4,512,1024)

  // Workspace carve-up (~190 MB total, all float32)
  float* ws   = (float*)d_ws;
  float* pe_v = ws;           ws += (size_t)VN * DDIM;
  float* pe_q = ws;           ws += (size_t)QN * DDIM;
  float* v_   = ws;           ws += (size_t)BB * VN * HK;
  float* vv_  = ws;           ws += (size_t)BB * VN * HK;
  float* q_   = ws;           ws += (size_t)BB * QN * HK;
  float* vq_  = ws;           ws += (size_t)BB * QN * HK;
  float* Ab   = ws;           ws += (size_t)BB * HK * HK;
  float* Bb   = ws;           ws += (size_t)BB * HK * HK;
  float* Sv   = ws;           ws += (size_t)BB * HK;
  float* Sq   = ws;           ws += (size_t)BB * HK;
  float* Hs   = ws;           ws += HK;
  float* hbs  = ws;           ws += 1;
  float* Lk   = ws;           ws += (size_t)BB * HK;

  // 1) PE tables
  pe_kernel<<<VN, DDIM, 0, stream>>>(pe_v, VN);
  pe_kernel<<<QN, DDIM, 0, stream>>>(pe_q, QN);

  // 2) Input GEMMs (fused PE+raw branches):  v-side then q-side
  //    v: Mtot=16384 -> 1024 m-tiles * 12 n-tiles / 8 waves = 1536 blocks
  prep_kernel<<<1536, 256, 0, stream>>>(v, pe_v, Wv, bv, Wvv, bvv, v_, vv_, VN);
  //    q: Mtot=32768 -> 2048 m-tiles * 12 n-tiles / 8 waves = 3072 blocks
  prep_kernel<<<3072, 256, 0, stream>>>(q, pe_q, Wq, bq, Wvq, bvq, q_, vq_, QN);

  // 3) att_maps (dominant GEMM, writes straight to d_out):
  //    32*4*(512/16)*(1024/64) = 65536 wave-tiles / 8 = 8192 blocks
  att_kernel<<<8192, 256, 0, stream>>>(v_, q_, h_mat, h_bias, out_att);

  // 4) Gram matrices for the factorized trilinear pooling
  //    32*24*6 = 4608 wave-tiles / 8 = 576 blocks each
  corr_kernel<<<576, 256, 0, stream>>>(vv_, v_, Ab, VN);
  corr_kernel<<<576, 256, 0, stream>>>(vq_, q_, Bb, QN);

  // 5) Reductions for the bias term + h_mat head-sum
  colsum_kernel<<<(BB * HK + 255) / 256, 256, 0, stream>>>(vv_, Sv, VN);
  colsum_kernel<<<(BB * HK + 255) / 256, 256, 0, stream>>>(vq_, Sq, QN);
  hsum_kernel<<<1, HK, 0, stream>>>(h_mat, h_bias, Hs, hbs);

  // 6) logits_k, then K-group sum + BatchNorm
  logitsk_kernel<<<BB * HK, 128, 0, stream>>>(Ab, Bb, Hs, hbs, Sv, Sq, Lk);
  final_kernel<<<HD, 32, 0, stream>>>(Lk, gamma, beta, out_logits);
}